// BiMpmMatching_10565619549068
// MI455X (gfx1250) — compile-verified
//
#include <hip/hip_runtime.h>
#include <hip/hip_bf16.h>
#include <math.h>

// ---------------- problem constants (from reference) ----------------
constexpr int B  = 16;
constexpr int S1 = 96;
constexpr int S2 = 112;
constexpr int H  = 384;
constexpr int P  = 24;
constexpr int C  = 5 + 5 * P;      // 125 output columns per direction

constexpr float TINYF  = 1e-13f;
constexpr float EPSF   = 1e-8f;
constexpr float BIGNEG = -1e30f;

// ---------------- workspace layout (floats) ----------------
constexpr size_t OFF_C1M  = 0;                               // (B,S1,H) masked context_1
constexpr size_t OFF_C2M  = OFF_C1M  + (size_t)B*S1*H;       // (B,S2,H) masked context_2
constexpr size_t OFF_N1   = OFF_C2M  + (size_t)B*S2*H;       // (B,S1)   clamped norms
constexpr size_t OFF_N2   = OFF_N1   + (size_t)B*S1;         // (B,S2)
constexpr size_t OFF_NMP1 = OFF_N2   + (size_t)B*S2;         // (B,P,S1) weighted norms (maxpool)
constexpr size_t OFF_NMP2 = OFF_NMP1 + (size_t)B*P*S1;       // (B,P,S2)
constexpr size_t OFF_COS  = OFF_NMP2 + (size_t)B*P*S2;       // (B,S1,S2)
constexpr size_t OFF_RS   = OFF_COS  + (size_t)B*S1*S2;      // (B,S1) full row sums of cos
constexpr size_t OFF_CS   = OFF_RS   + (size_t)B*S1;         // (B,S2) full col sums of cos
constexpr size_t OFF_AM2  = OFF_CS   + (size_t)B*S2;         // (B,S1,H) att_mean_2
constexpr size_t OFF_AM1  = OFF_AM2  + (size_t)B*S1*H;       // (B,S2,H) att_mean_1
constexpr size_t OFF_AX2  = OFF_AM1  + (size_t)B*S2*H;       // (B,S1,H) att_max_2
constexpr size_t OFF_AX1  = OFF_AX2  + (size_t)B*S1*H;       // (B,S2,H) att_max_1
constexpr size_t OFF_L1   = OFF_AX1  + (size_t)B*S2*H;       // (B,H)    c1_last
constexpr size_t OFF_L2   = OFF_L1   + (size_t)B*H;          // (B,H)    c2_last
constexpr size_t OFF_MP   = OFF_L2   + (size_t)B*H;          // (B,P,S1,S2) pairwise sims

// ---------------- WMMA types / helper ----------------
typedef __attribute__((ext_vector_type(2))) float v2f;
typedef __attribute__((ext_vector_type(8))) float v8f;

__device__ __forceinline__ v8f wmma_k4(v2f a, v2f b, v8f c) {
    // D = A(16x4 f32) * B(4x16 f32) + C(16x16 f32)
    return __builtin_amdgcn_wmma_f32_16x16x4_f32(
        /*neg_a=*/false, a, /*neg_b=*/false, b,
        /*c_mod=*/(short)0, c, /*reuse_a=*/false, /*reuse_b=*/false);
}

// =====================================================================
// Kernel 1: mask contexts, unweighted norms (EPS clamp), per-p weighted
// norms for maxpool match.  One 128-thread block per sequence row.
// =====================================================================
__global__ __launch_bounds__(128) void prep_kernel(
    const float* __restrict__ ctx1, const unsigned char* __restrict__ mask1,
    const float* __restrict__ ctx2, const unsigned char* __restrict__ mask2,
    const float* __restrict__ w_mp, float* __restrict__ ws) {
    __shared__ float row[H];
    __shared__ float red[128];
    const int r = blockIdx.x, tid = threadIdx.x;
    const float* src; float* dst; float* nrm; float* nmp;
    const unsigned char* mask;
    int b, s, S;
    if (r < B * S1) {
        b = r / S1; s = r % S1; S = S1; src = ctx1; mask = mask1;
        dst = ws + OFF_C1M; nrm = ws + OFF_N1; nmp = ws + OFF_NMP1;
    } else {
        const int r2 = r - B * S1;
        b = r2 / S2; s = r2 % S2; S = S2; src = ctx2; mask = mask2;
        dst = ws + OFF_C2M; nrm = ws + OFF_N2; nmp = ws + OFF_NMP2;
    }
    const float mval = mask[b * S + s] ? 1.0f : 0.0f;
    const size_t base = ((size_t)b * S + s) * H;
    float ss = 0.0f;
    for (int h = tid; h < H; h += 128) {
        const float v = src[base + h] * mval;
        row[h] = v; dst[base + h] = v; ss += v * v;
    }
    red[tid] = ss; __syncthreads();
    for (int off = 64; off > 0; off >>= 1) {
        if (tid < off) red[tid] += red[tid + off];
        __syncthreads();
    }
    if (tid == 0) nrm[b * S + s] = fmaxf(sqrtf(red[0]), EPSF);
    __syncthreads();
    for (int p = 0; p < P; ++p) {
        float acc = 0.0f;
        for (int h = tid; h < H; h += 128) {
            const float w = w_mp[p * H + h];
            const float v = row[h];
            acc += w * w * v * v;
        }
        red[tid] = acc; __syncthreads();
        for (int off = 64; off > 0; off >>= 1) {
            if (tid < off) red[tid] += red[tid + off];
            __syncthreads();
        }
        if (tid == 0) nmp[(b * P + p) * S + s] = sqrtf(red[0]);
        __syncthreads();
    }
}

// =====================================================================
// Kernel 2: gather last unmasked row of each masked context.
// =====================================================================
__global__ __launch_bounds__(128) void lastvec_kernel(
    const unsigned char* __restrict__ mask1, const unsigned char* __restrict__ mask2,
    float* __restrict__ ws) {
    const int b = blockIdx.x >> 1, dir = blockIdx.x & 1;
    const int S = dir ? S2 : S1;
    const unsigned char* mask = dir ? mask2 : mask1;
    const float* cm = ws + (dir ? OFF_C2M : OFF_C1M);
    float* out = ws + (dir ? OFF_L2 : OFF_L1);
    int cnt = 0;
    for (int s = 0; s < S; ++s) cnt += mask[b * S + s] ? 1 : 0;
    const int idx = (cnt > 0) ? (cnt - 1) : 0;
    for (int h = threadIdx.x; h < H; h += 128)
        out[(size_t)b * H + h] = cm[((size_t)b * S + idx) * H + h];
}

// =====================================================================
// Kernel 3: cosine matrix via WMMA f32 16x16x4.  One wave per 16x16 tile.
// =====================================================================
__global__ __launch_bounds__(32) void cos_gemm_kernel(float* __restrict__ ws) {
    constexpr int T1 = S1 / 16, T2 = S2 / 16;
    const int id = blockIdx.x;
    const int b  = id / (T1 * T2);
    const int t  = id % (T1 * T2);
    const int i0 = (t / T2) * 16, j0 = (t % T2) * 16;
    const int lane = threadIdx.x;
    const int r  = lane & 15;
    const int kk = (lane >> 4) * 2;
    const float* c1m = ws + OFF_C1M;
    const float* c2m = ws + OFF_C2M;
    const float* arow = c1m + ((size_t)b * S1 + i0 + r) * H + kk;
    const float* brow = c2m + ((size_t)b * S2 + j0 + r) * H + kk;
    v8f acc = {};
    for (int k = 0; k < H; k += 4) {
        const v2f a  = *(const v2f*)(arow + k);
        const v2f bm = *(const v2f*)(brow + k);
        acc = wmma_k4(a, bm, acc);
    }
    const float* n1 = ws + OFF_N1;
    const float* n2 = ws + OFF_N2;
    float* cosm = ws + OFF_COS;
    const int n  = r;
    const int mh = (lane >> 4) * 8;
    const float n2v = n2[b * S2 + j0 + n];
    for (int v = 0; v < 8; ++v) {
        const int m = mh + v;
        const float n1v = n1[b * S1 + i0 + m];
        cosm[((size_t)b * S1 + i0 + m) * S2 + j0 + n] = acc[v] / (n1v * n2v);
    }
}

// =====================================================================
// Kernel 4: masked max/mean of cos (output cols 0,1) + full row/col sums.
// =====================================================================
__global__ void reduce_cos_kernel(
    const unsigned char* __restrict__ mask1, const unsigned char* __restrict__ mask2,
    float* __restrict__ ws, float* __restrict__ out) {
    const int idx = blockIdx.x * blockDim.x + threadIdx.x;
    const float* cosm = ws + OFF_COS;
    const int NR1 = B * S1;
    if (idx < NR1) {
        const int b = idx / S1, i = idx % S1;
        const float* row = cosm + ((size_t)b * S1 + i) * S2;
        float mx = BIGNEG, sm = 0.0f, full = 0.0f; int cnt = 0;
        for (int j = 0; j < S2; ++j) {
            const float v = row[j]; full += v;
            if (mask2[b * S2 + j]) { mx = fmaxf(mx, v); sm += v; ++cnt; }
        }
        float* mv1 = out + ((size_t)b * S1 + i) * C;
        mv1[0] = mx;
        mv1[1] = sm / fmaxf((float)cnt, TINYF);
        ws[OFF_RS + b * S1 + i] = full;
    } else if (idx < NR1 + B * S2) {
        const int r = idx - NR1;
        const int b = r / S2, j = r % S2;
        float mx = BIGNEG, sm = 0.0f, full = 0.0f; int cnt = 0;
        for (int i = 0; i < S1; ++i) {
            const float v = cosm[((size_t)b * S1 + i) * S2 + j]; full += v;
            if (mask1[b * S1 + i]) { mx = fmaxf(mx, v); sm += v; ++cnt; }
        }
        float* mv2 = out + (size_t)B * S1 * C + ((size_t)b * S2 + j) * C;
        mv2[0] = mx;
        mv2[1] = sm / fmaxf((float)cnt, TINYF);
        ws[OFF_CS + b * S2 + j] = full;
    }
}

// =====================================================================
// Kernel 5a: att_mean_2 = (cos @ c2m) / clamp(rowsum)   (WMMA, K = S2)
// =====================================================================
__global__ __launch_bounds__(32) void att2_gemm_kernel(float* __restrict__ ws) {
    constexpr int T1 = S1 / 16, TH = H / 16;
    const int id = blockIdx.x;
    const int b  = id / (T1 * TH);
    const int t  = id % (T1 * TH);
    const int i0 = (t / TH) * 16, n0 = (t % TH) * 16;
    const int lane = threadIdx.x;
    const int r  = lane & 15;
    const int kk = (lane >> 4) * 2;
    const float* cosm = ws + OFF_COS;
    const float* c2m  = ws + OFF_C2M;
    const float* arow = cosm + ((size_t)b * S1 + i0 + r) * S2 + kk;
    v8f acc = {};
    for (int k = 0; k < S2; k += 4) {
        const v2f a = *(const v2f*)(arow + k);
        v2f bm;
        bm.x = c2m[((size_t)b * S2 + k + kk    ) * H + n0 + r];
        bm.y = c2m[((size_t)b * S2 + k + kk + 1) * H + n0 + r];
        acc = wmma_k4(a, bm, acc);
    }
    const float* rs = ws + OFF_RS;
    float* att = ws + OFF_AM2;
    const int n = r, mh = (lane >> 4) * 8;
    for (int v = 0; v < 8; ++v) {
        const int m = mh + v;
        const float d = fmaxf(rs[b * S1 + i0 + m], TINYF);
        att[((size_t)b * S1 + i0 + m) * H + n0 + n] = acc[v] / d;
    }
}

// =====================================================================
// Kernel 5b: att_mean_1 = (cos^T @ c1m) / clamp(colsum)  (WMMA, K = S1)
// =====================================================================
__global__ __launch_bounds__(32) void att1_gemm_kernel(float* __restrict__ ws) {
    constexpr int T2 = S2 / 16, TH = H / 16;
    const int id = blockIdx.x;
    const int b  = id / (T2 * TH);
    const int t  = id % (T2 * TH);
    const int j0 = (t / TH) * 16, n0 = (t % TH) * 16;
    const int lane = threadIdx.x;
    const int r  = lane & 15;
    const int kk = (lane >> 4) * 2;
    const float* cosm = ws + OFF_COS;
    const float* c1m  = ws + OFF_C1M;
    v8f acc = {};
    for (int k = 0; k < S1; k += 4) {
        v2f a, bm;
        a.x  = cosm[((size_t)b * S1 + k + kk    ) * S2 + j0 + r];
        a.y  = cosm[((size_t)b * S1 + k + kk + 1) * S2 + j0 + r];
        bm.x = c1m[((size_t)b * S1 + k + kk    ) * H + n0 + r];
        bm.y = c1m[((size_t)b * S1 + k + kk + 1) * H + n0 + r];
        acc = wmma_k4(a, bm, acc);
    }
    const float* cs = ws + OFF_CS;
    float* att = ws + OFF_AM1;
    const int n = r, mh = (lane >> 4) * 8;
    for (int v = 0; v < 8; ++v) {
        const int m = mh + v;
        const float d = fmaxf(cs[b * S2 + j0 + m], TINYF);
        att[((size_t)b * S2 + j0 + m) * H + n0 + n] = acc[v] / d;
    }
}

// =====================================================================
// Kernel 6: max-attentive vectors (masked max of cos[i,j]*c[j,h]).
// =====================================================================
__global__ __launch_bounds__(128) void attmax_kernel(
    const unsigned char* __restrict__ mask1, const unsigned char* __restrict__ mask2,
    float* __restrict__ ws) {
    __shared__ float cs[S2 > S1 ? S2 : S1];
    __shared__ unsigned char msk[S2 > S1 ? S2 : S1];
    const int r = blockIdx.x, tid = threadIdx.x;
    const float* cosm = ws + OFF_COS;
    if (r < B * S1) {
        const int b = r / S1, i = r % S1;
        for (int j = tid; j < S2; j += 128) {
            cs[j]  = cosm[((size_t)b * S1 + i) * S2 + j];
            msk[j] = mask2[b * S2 + j];
        }
        __syncthreads();
        const float* c2m = ws + OFF_C2M;
        float* outp = ws + OFF_AX2;
        for (int h = tid; h < H; h += 128) {
            float m = BIGNEG;
            for (int j = 0; j < S2; ++j)
                if (msk[j]) m = fmaxf(m, cs[j] * c2m[((size_t)b * S2 + j) * H + h]);
            outp[((size_t)b * S1 + i) * H + h] = m;
        }
    } else {
        const int r2 = r - B * S1;
        const int b = r2 / S2, j = r2 % S2;
        for (int i = tid; i < S1; i += 128) {
            cs[i]  = cosm[((size_t)b * S1 + i) * S2 + j];
            msk[i] = mask1[b * S1 + i];
        }
        __syncthreads();
        const float* c1m = ws + OFF_C1M;
        float* outp = ws + OFF_AX1;
        for (int h = tid; h < H; h += 128) {
            float m = BIGNEG;
            for (int i = 0; i < S1; ++i)
                if (msk[i]) m = fmaxf(m, cs[i] * c1m[((size_t)b * S1 + i) * H + h]);
            outp[((size_t)b * S2 + j) * H + h] = m;
        }
    }
}

// =====================================================================
// Kernel 7: maxpool pairwise match — the dominant GEMM family.
// One wave per (b, p, 16x16 tile).  w_p^2 folded into A operand.
// =====================================================================
__global__ __launch_bounds__(32) void mp_gemm_kernel(
    const float* __restrict__ w_mp, float* __restrict__ ws) {
    constexpr int T1 = S1 / 16, T2 = S2 / 16;
    const int id = blockIdx.x;
    const int bp = id / (T1 * T2);
    const int t  = id % (T1 * T2);
    const int b = bp / P, p = bp % P;
    const int i0 = (t / T2) * 16, j0 = (t % T2) * 16;
    const int lane = threadIdx.x;
    const int r  = lane & 15;
    const int kk = (lane >> 4) * 2;
    const float* c1m = ws + OFF_C1M;
    const float* c2m = ws + OFF_C2M;
    const float* arow = c1m + ((size_t)b * S1 + i0 + r) * H + kk;
    const float* brow = c2m + ((size_t)b * S2 + j0 + r) * H + kk;
    const float* wrow = w_mp + p * H + kk;
    v8f acc = {};
    for (int k = 0; k < H; k += 4) {
        v2f a = *(const v2f*)(arow + k);
        const v2f wv = *(const v2f*)(wrow + k);
        a.x *= wv.x * wv.x;
        a.y *= wv.y * wv.y;
        const v2f bm = *(const v2f*)(brow + k);
        acc = wmma_k4(a, bm, acc);
    }
    const float* na = ws + OFF_NMP1;
    const float* nb = ws + OFF_NMP2;
    float* mp = ws + OFF_MP;
    const int n = r, mh = (lane >> 4) * 8;
    const float nbv = nb[(b * P + p) * S2 + j0 + n];
    for (int v = 0; v < 8; ++v) {
        const int m = mh + v;
        const float nav = na[(b * P + p) * S1 + i0 + m];
        mp[(((size_t)(b * P + p)) * S1 + i0 + m) * S2 + j0 + n] =
            acc[v] / fmaxf(nav * nbv, TINYF);
    }
}

// =====================================================================
// Kernel 8: masked max/mean of pairwise sims (output cols 27..74).
// =====================================================================
__global__ void reduce_mp_kernel(
    const unsigned char* __restrict__ mask1, const unsigned char* __restrict__ mask2,
    const float* __restrict__ ws, float* __restrict__ out) {
    const int idx = blockIdx.x * blockDim.x + threadIdx.x;
    const float* mp = ws + OFF_MP;
    const int N1 = B * S1 * P;
    if (idx < N1) {
        const int p = idx % P;
        const int rest = idx / P;
        const int i = rest % S1, b = rest / S1;
        const float* row = mp + (((size_t)(b * P + p)) * S1 + i) * S2;
        float mx = BIGNEG, sm = 0.0f; int cnt = 0;
        for (int j = 0; j < S2; ++j)
            if (mask2[b * S2 + j]) { const float v = row[j]; mx = fmaxf(mx, v); sm += v; ++cnt; }
        float* mv1 = out + ((size_t)b * S1 + i) * C;
        mv1[27 + p] = mx;
        mv1[51 + p] = sm / fmaxf((float)cnt, TINYF);
    } else if (idx < N1 + B * S2 * P) {
        const int r = idx - N1;
        const int p = r % P;
        const int rest = r / P;
        const int j = rest % S2, b = rest / S2;
        float mx = BIGNEG, sm = 0.0f; int cnt = 0;
        for (int i = 0; i < S1; ++i)
            if (mask1[b * S1 + i]) {
                const float v = mp[(((size_t)(b * P + p)) * S1 + i) * S2 + j];
                mx = fmaxf(mx, v); sm += v; ++cnt;
            }
        float* mv2 = out + (size_t)B * S1 * C + ((size_t)b * S2 + j) * C;
        mv2[27 + p] = mx;
        mv2[51 + p] = sm / fmaxf((float)cnt, TINYF);
    }
}

// =====================================================================
// Kernel 9: mp_match for (full, attentive, max-attentive) pairs.
// One thread per (direction-row, pair, slot); slot==P -> sim_single.
// =====================================================================
__global__ void match_kernel(
    const float* __restrict__ w_full, const float* __restrict__ w_att,
    const float* __restrict__ w_max_att, const float* __restrict__ ws,
    float* __restrict__ out) {
    constexpr int SLOTS = P + 1;          // 25
    constexpr int PER_ROW = 3 * SLOTS;    // 75
    const int idx = blockIdx.x * blockDim.x + threadIdx.x;
    const int total = B * (S1 + S2) * PER_ROW;
    if (idx >= total) return;
    const int sp   = idx % PER_ROW;
    const int row  = idx / PER_ROW;
    const int pair = sp / SLOTS;
    const int slot = sp % SLOTS;
    int dir, b, s, S;
    if (row < B * S1) { dir = 0; b = row / S1; s = row % S1; S = S1; }
    else { const int r2 = row - B * S1; dir = 1; b = r2 / S2; s = r2 % S2; S = S2; }
    const float* v1 = ws + (dir ? OFF_C2M : OFF_C1M) + ((size_t)b * S + s) * H;
    const float* v2;
    const float* W;
    int colbase;
    if (pair == 0)      { v2 = ws + (dir ? OFF_L1  : OFF_L2 ) + (size_t)b * H;           W = w_full;    colbase = 2;   }
    else if (pair == 1) { v2 = ws + (dir ? OFF_AM1 : OFF_AM2) + ((size_t)b * S + s) * H; W = w_att;     colbase = 75;  }
    else                { v2 = ws + (dir ? OFF_AX1 : OFF_AX2) + ((size_t)b * S + s) * H; W = w_max_att; colbase = 100; }
    float* outrow = out + (dir ? (size_t)B * S1 * C : 0) + ((size_t)b * S + s) * C;
    float d = 0.0f, a = 0.0f, bb = 0.0f;
    if (slot == P) {
        for (int h = 0; h < H; ++h) {
            const float x = v1[h], y = v2[h];
            d += x * y; a += x * x; bb += y * y;
        }
        outrow[colbase] = d / (fmaxf(sqrtf(a), EPSF) * fmaxf(sqrtf(bb), EPSF));
    } else {
        const float* wr = W + slot * H;
        for (int h = 0; h < H; ++h) {
            const float w = wr[h];
            const float w2 = w * w;
            const float x = v1[h], y = v2[h];
            d += w2 * x * y; a += w2 * x * x; bb += w2 * y * y;
        }
        outrow[colbase + 1 + slot] = d / (fmaxf(sqrtf(a), EPSF) * fmaxf(sqrtf(bb), EPSF));
    }
}

// =====================================================================
// launch
// =====================================================================
extern "C" void kernel_launch(void* const* d_in, const int* in_sizes, int n_in,
                              void* d_out, int out_size, void* d_ws, size_t ws_size,
                              hipStream_t stream) {
    (void)in_sizes; (void)n_in; (void)out_size; (void)ws_size;
    const float*         ctx1     = (const float*)d_in[0];
    const unsigned char* mask1    = (const unsigned char*)d_in[1];   // jnp bool -> raw bytes
    const float*         ctx2     = (const float*)d_in[2];
    const unsigned char* mask2    = (const unsigned char*)d_in[3];
    const float*         w_full   = (const float*)d_in[4];
    const float*         w_mp     = (const float*)d_in[5];
    const float*         w_att    = (const float*)d_in[6];
    const float*         w_maxatt = (const float*)d_in[7];
    float* out = (float*)d_out;
    float* ws  = (float*)d_ws;

    prep_kernel<<<B * (S1 + S2), 128, 0, stream>>>(ctx1, mask1, ctx2, mask2, w_mp, ws);
    lastvec_kernel<<<2 * B, 128, 0, stream>>>(mask1, mask2, ws);
    cos_gemm_kernel<<<B * (S1 / 16) * (S2 / 16), 32, 0, stream>>>(ws);
    reduce_cos_kernel<<<(B * (S1 + S2) + 255) / 256, 256, 0, stream>>>(mask1, mask2, ws, out);
    att2_gemm_kernel<<<B * (S1 / 16) * (H / 16), 32, 0, stream>>>(ws);
    att1_gemm_kernel<<<B * (S2 / 16) * (H / 16), 32, 0, stream>>>(ws);
    attmax_kernel<<<B * (S1 + S2), 128, 0, stream>>>(mask1, mask2, ws);
    mp_gemm_kernel<<<B * P * (S1 / 16) * (S2 / 16), 32, 0, stream>>>(w_mp, ws);
    reduce_mp_kernel<<<(B * (S1 + S2) * P + 255) / 256, 256, 0, stream>>>(mask1, mask2, ws, out);
    match_kernel<<<(B * (S1 + S2) * 3 * (P + 1) + 255) / 256, 256, 0, stream>>>(
        w_full, w_att, w_maxatt, ws, out);
}